// PositionEmbeddingIPM_80161269612759
// MI455X (gfx1250) — compile-verified
//
#include <hip/hip_runtime.h>
#include <hip/hip_bf16.h>
#include <math.h>

typedef float v2f __attribute__((ext_vector_type(2)));
typedef float v8f __attribute__((ext_vector_type(8)));

#define BATCH   8
#define N_CAM   6
#define H_OUT   64
#define W_CAM   224
#define NW      (N_CAM * W_CAM)     // 1344
#define NPF     256
#define CH_STRIDE ((size_t)H_OUT * NW)  // 86016 floats between channels

// ---------------------------------------------------------------------------
// Stage 1: per (b, cam) precompute RM = R @ K^{-1} @ ida^{-1} and t_z.
// 48 tiny 3x3 inversions -> d_ws (12 floats per camera, padded).
// ---------------------------------------------------------------------------
__device__ inline void inv3x3(const float m[9], float inv[9]) {
    float a = m[0], b = m[1], c = m[2];
    float d = m[3], e = m[4], f = m[5];
    float g = m[6], h = m[7], i = m[8];
    float A =  (e * i - f * h);
    float B = -(d * i - f * g);
    float C =  (d * h - e * g);
    float det = a * A + b * B + c * C;
    float rd = 1.0f / det;
    inv[0] = A * rd;              inv[1] = -(b * i - c * h) * rd;  inv[2] =  (b * f - c * e) * rd;
    inv[3] = B * rd;              inv[4] =  (a * i - c * g) * rd;  inv[5] = -(a * f - c * d) * rd;
    inv[6] = C * rd;              inv[7] = -(a * h - b * g) * rd;  inv[8] =  (a * e - b * d) * rd;
}

__global__ void ipm_setup_kernel(const float* __restrict__ extr,   // (8,6,4,4)
                                 const float* __restrict__ intr,   // (8,6,3,3)
                                 const float* __restrict__ ida,    // (8,6,3,3)
                                 float* __restrict__ rm_tz)        // (48,12)
{
    int idx = blockIdx.x * blockDim.x + threadIdx.x;
    if (idx >= BATCH * N_CAM) return;

    float K[9], A[9], invK[9], invA[9], M[9], RM[9];
    const float* Kp = intr + idx * 9;
    const float* Ap = ida  + idx * 9;
    const float* Ep = extr + idx * 16;
    for (int i = 0; i < 9; ++i) { K[i] = Kp[i]; A[i] = Ap[i]; }
    inv3x3(K, invK);
    inv3x3(A, invA);
    // M = invK @ invA
    for (int r = 0; r < 3; ++r)
        for (int c = 0; c < 3; ++c)
            M[r * 3 + c] = invK[r * 3 + 0] * invA[0 * 3 + c]
                         + invK[r * 3 + 1] * invA[1 * 3 + c]
                         + invK[r * 3 + 2] * invA[2 * 3 + c];
    // RM = R @ M  (R = extrinsic[:3,:3], extrinsic row-major 4x4)
    for (int r = 0; r < 3; ++r)
        for (int c = 0; c < 3; ++c)
            RM[r * 3 + c] = Ep[r * 4 + 0] * M[0 * 3 + c]
                          + Ep[r * 4 + 1] * M[1 * 3 + c]
                          + Ep[r * 4 + 2] * M[2 * 3 + c];

    float* o = rm_tz + idx * 12;
    for (int i = 0; i < 9; ++i) o[i] = RM[i];
    o[9] = Ep[2 * 4 + 3];   // t_z = extrinsic[2][3]
}

// ---------------------------------------------------------------------------
// Stage 2: one wave per 16-pixel group.
//   WMMA f32 16x16x4:  A = RM rows (0..2 dup'd into 8..10), B = coord columns.
//   -> lane l holds world {x,y,z} of pixel l%16 in c[0..2]; lanes 16..31 are
//      the duplicate, used to emit the second half of the channel space.
// ---------------------------------------------------------------------------
__global__ void __launch_bounds__(256) ipm_main_kernel(
    const float* __restrict__ rm_tz,
    const int*   __restrict__ do_flip,
    float* __restrict__ pos,     // (8, 256, 64, 1344)
    float* __restrict__ valid)   // (8, 64, 1344)
{
    const int lane  = threadIdx.x & 31;
    const int gwave = blockIdx.x * (blockDim.x >> 5) + (threadIdx.x >> 5);

    const int groupsPerRow = NW / 16;              // 84
    const int wgGroup = gwave % groupsPerRow;
    int rem = gwave / groupsPerRow;
    const int h = rem % H_OUT;
    const int b = rem / H_OUT;

    const int pl = lane & 15;                      // pixel within group
    const int hi = lane >> 4;                      // 0: y-half, 1: x-half
    const int wg = wgGroup * 16 + pl;              // stitched width index
    const int cam = wg / W_CAM;
    const int w   = wg - cam * W_CAM;

    const float* P = rm_tz + (b * N_CAM + cam) * 12;
    const float tz = P[9];

    // pixel-grid coords: linspace(0, IN-1, OUT)
    const float xc = (float)w * (895.0f / 223.0f);
    const float yc = (float)h * (511.0f / 63.0f);

    // ---- A operand (16x4): rows 0..2 and 8..10 carry RM, rest zero ----
    const int m = pl;
    const int r = (m < 3) ? m : ((m >= 8 && m < 11) ? (m - 8) : -1);
    v2f a; a.x = 0.0f; a.y = 0.0f;
    if (r >= 0) {
        a.x = (hi == 0) ? P[r * 3 + 0] : P[r * 3 + 2];
        a.y = (hi == 0) ? P[r * 3 + 1] : 0.0f;
    }
    // ---- B operand (4x16): column n = [x_n, y_n, 1, 0]^T ----
    v2f bm;
    bm.x = (hi == 0) ? xc : 1.0f;
    bm.y = (hi == 0) ? yc : 0.0f;

    v8f c = {};
    c = __builtin_amdgcn_wmma_f32_16x16x4_f32(
            /*neg_a=*/false, a, /*neg_b=*/false, bm,
            /*c_mod=*/(short)0, c, /*reuse_a=*/false, /*reuse_b=*/false);

    const float wx = c[0], wy = c[1], wz = c[2];

    // F.normalize (eps-clamped), depth, validity, xy
    float nrm = sqrtf(wx * wx + wy * wy + wz * wz);
    nrm = fmaxf(nrm, 1e-12f);
    const float inv   = 1.0f / nrm;
    const float z     = wz * inv;
    const float depth = -tz / z;
    const bool  vld   = depth > 0.0f;
    float xcomp = wx * inv * depth;
    float ycomp = wy * inv * depth;
    if (do_flip[0]) ycomp = -ycomp;

    // channels 0..127 encode y (py), 128..255 encode x (px)
    const float val = (hi == 0) ? ycomp : xcomp;

    // pos offset of [b][0][h][wg]
    const size_t base0 = ((size_t)b * NPF * H_OUT + (size_t)h) * (size_t)NW + (size_t)wg;
    float* outBase = pos + base0 + (size_t)(hi * 128) * CH_STRIDE;

    // dim_t pairs: t_k = 10000^(k/64); log2(10000)/64 = 0.2076205059...
    const float kStep = 0.20762050593046f;
    #pragma unroll 4
    for (int k = 0; k < 64; ++k) {
        const float arg = val * exp2f(-(float)k * kStep);
        float s, co;
        __sincosf(arg, &s, &co);
        outBase[(size_t)(2 * k)     * CH_STRIDE] = vld ? s  : 0.0f;
        outBase[(size_t)(2 * k + 1) * CH_STRIDE] = vld ? co : 0.0f;
    }

    if (hi == 0) {
        valid[((size_t)b * H_OUT + (size_t)h) * NW + wg] = vld ? 1.0f : 0.0f;
    }
}

// ---------------------------------------------------------------------------
extern "C" void kernel_launch(void* const* d_in, const int* in_sizes, int n_in,
                              void* d_out, int out_size, void* d_ws, size_t ws_size,
                              hipStream_t stream) {
    const float* extr = (const float*)d_in[0];   // (8,6,4,4)
    const float* intr = (const float*)d_in[1];   // (8,6,3,3)
    const float* ida  = (const float*)d_in[2];   // (8,6,3,3)
    const int*   flip = (const int*)d_in[3];     // scalar

    float* pos   = (float*)d_out;
    float* valid = pos + (size_t)BATCH * NPF * H_OUT * NW;   // 176,160,768
    float* rm_tz = (float*)d_ws;                             // 48*12 floats

    ipm_setup_kernel<<<1, 64, 0, stream>>>(extr, intr, ida, rm_tz);

    // 8*64*84 = 43008 waves, 8 waves/block -> 5376 blocks, no partial waves
    ipm_main_kernel<<<5376, 256, 0, stream>>>(rm_tz, flip, pos, valid);
}